// WaveLMv6_52948356825491
// MI455X (gfx1250) — compile-verified
//
#include <hip/hip_runtime.h>
#include <math.h>

#define NV 8192
#define NH 7
#define NH2 14
#define NB 2
#define NT 64

// ---------------------------------------------------------------------------
// Kernel 1: cross terms  C[b,t,v] = 2*cos(phi_t - phi_v) * sum_ij sA_i cA_j sinc(2(sf_i-cf_j))
// (phase broadcast in the reference makes cos(dphi) independent of i,j)
// ---------------------------------------------------------------------------

__device__ __forceinline__ float sinc_ref(float x) {
    // jnp.sinc(x) = sin(pi*x)/(pi*x), sinc(0)=1
    float px = 3.14159265358979323846f * x;
    float s = __sinf(px) / px;
    return (fabsf(px) < 1e-8f) ? 1.0f : s;
}

__global__ void WaveLM_cross_kernel(const int* __restrict__ ids,
                                    const float* __restrict__ fslow,
                                    const float* __restrict__ ffast,
                                    const float* __restrict__ amp,
                                    const float* __restrict__ ph,
                                    const float* __restrict__ smix,
                                    const float* __restrict__ dslow_p,
                                    const float* __restrict__ dfast_p,
                                    float* __restrict__ Cbuf) {
    const int v = blockIdx.x * blockDim.x + threadIdx.x;   // 0..NV-1
    const int t = blockIdx.y;
    const int b = blockIdx.z;
    const float dslow = dslow_p[0];
    const float dfast = dfast_p[0];

    const int id = ids[b * NT + t];                        // uniform per block
    const float tf_s = fslow[id], tf_f = ffast[id];
    const float tA   = amp[id],   tphi = ph[id];
    const float tmix = 1.0f / (1.0f + __expf(-smix[id]));

    const float cf_s = fslow[v],  cf_f = ffast[v];
    const float cA0  = amp[v],    cphi = ph[v];
    const float cmix = 1.0f / (1.0f + __expf(-smix[v]));

    float sf[NH2], sA[NH2], cf[NH2], cA[NH2];
#pragma unroll
    for (int a = 0; a < NH; ++a) {
        const float h  = (float)(a + 1);
        const float hs = __powf(h, dslow);
        const float hf = __powf(h, dfast);
        sf[a]        = tf_s * h;  sA[a]        = (tmix * tA) / hs;
        sf[a + NH]   = tf_f * h;  sA[a + NH]   = ((1.0f - tmix) * tA) / hf;
        cf[a]        = cf_s * h;  cA[a]        = (cmix * cA0) / hs;
        cf[a + NH]   = cf_f * h;  cA[a + NH]   = ((1.0f - cmix) * cA0) / hf;
    }

    float S = 0.0f;
#pragma unroll
    for (int i = 0; i < NH2; ++i) {
        float acc = 0.0f;
#pragma unroll
        for (int j = 0; j < NH2; ++j) {
            acc = fmaf(cA[j], sinc_ref(2.0f * (sf[i] - cf[j])), acc);
        }
        S = fmaf(sA[i], acc, S);
    }

    Cbuf[((size_t)b * NT + t) * NV + v] = 2.0f * __cosf(tphi - cphi) * S;
}

// ---------------------------------------------------------------------------
// Kernel 2: decayed causal scan as GEMM via WMMA f32.
//   logits[b] = W (64x64 lower-triangular decay * invTemp) @ C[b] (64x8192)
// W is staged in LDS once per block; each wave owns a full 64-row column
// stripe (four 16x16 t-tiles) so every B fragment load feeds 4 WMMAs.
// Tile index is wave-uniform -> EXEC all 1s at every WMMA.
// ---------------------------------------------------------------------------

typedef float v2f __attribute__((ext_vector_type(2)));
typedef float v8f __attribute__((ext_vector_type(8)));

__global__ void WaveLM_scan_wmma_kernel(const float* __restrict__ Cbuf,
                                        const float* __restrict__ ls_p,
                                        const float* __restrict__ lf_p,
                                        const float* __restrict__ temp_p,
                                        float* __restrict__ out) {
    __shared__ __align__(16) float Wlds[NT * NT];          // 16 KB of 320 KB/WGP

    const int tid = (int)threadIdx.x;

    // softplus(x) = log(1+exp(x));  decay = exp(-(lam_s+lam_f)/2)
    const float lam_s = logf(1.0f + __expf(ls_p[0]));
    const float lam_f = logf(1.0f + __expf(lf_p[0]));
    const float decay = __expf(-0.5f * (lam_s + lam_f));
    const float invT  = 1.0f / (fabsf(temp_p[0]) + 1e-6f);

    // Materialize W once per block: W[t][tp] = decay^(t-1-tp)*invT for tp<t else 0
    for (int idx = tid; idx < NT * NT; idx += (int)blockDim.x) {
        const int t  = idx >> 6;
        const int tp = idx & 63;
        const float w = __powf(decay, (float)(t - 1 - tp)) * invT;
        Wlds[idx] = (tp < t) ? w : 0.0f;
    }
    __syncthreads();

    const int wave = (int)((blockIdx.x * blockDim.x + tid) >> 5); // global wave id
    const int lane = tid & 31;
    const int m    = lane & 15;        // A row / B,C,D column within tile
    const int g    = lane >> 4;        // lane half-group

    const int VT = NV / 16;            // 512 column tiles
    const int vv = wave % VT;
    const int b  = wave / VT;          // grid sized exactly to NB*VT waves
    const int v0 = vv * 16;

    v8f acc0 = {}, acc1 = {}, acc2 = {}, acc3 = {};

#pragma unroll 4
    for (int k = 0; k < NT; k += 4) {
        const int k0 = k + 2 * g;      // vgpr0 -> K=2g, vgpr1 -> K=2g+1

        // B fragment (4x16, K x N): lane column n=m; rows k0, k0+1
        const float* crow = Cbuf + ((size_t)b * NT + k0) * NV + v0 + m;
        v2f bf;
        bf.x = crow[0];
        bf.y = crow[NV];

        // A fragments for the four 16-row tiles, from LDS (8B-aligned pairs)
        const float2 w0 = *(const float2*)&Wlds[( 0 + m) * NT + k0];
        const float2 w1 = *(const float2*)&Wlds[(16 + m) * NT + k0];
        const float2 w2 = *(const float2*)&Wlds[(32 + m) * NT + k0];
        const float2 w3 = *(const float2*)&Wlds[(48 + m) * NT + k0];
        v2f a0, a1, a2, a3;
        a0.x = w0.x; a0.y = w0.y;
        a1.x = w1.x; a1.y = w1.y;
        a2.x = w2.x; a2.y = w2.y;
        a3.x = w3.x; a3.y = w3.y;

        acc0 = __builtin_amdgcn_wmma_f32_16x16x4_f32(false, a0, false, bf, (short)0, acc0, false, false);
        acc1 = __builtin_amdgcn_wmma_f32_16x16x4_f32(false, a1, false, bf, (short)0, acc1, false, false);
        acc2 = __builtin_amdgcn_wmma_f32_16x16x4_f32(false, a2, false, bf, (short)0, acc2, false, false);
        acc3 = __builtin_amdgcn_wmma_f32_16x16x4_f32(false, a3, false, bf, (short)0, acc3, false, false);
    }

    // D layout: VGPR r holds row M = r + 8*g, lane column n = m
    const int v = v0 + m;
#pragma unroll
    for (int r = 0; r < 8; ++r) {
        const int tr = r + 8 * g;
        out[((size_t)b * NT + ( 0 + tr)) * NV + v] = acc0[r];
        out[((size_t)b * NT + (16 + tr)) * NV + v] = acc1[r];
        out[((size_t)b * NT + (32 + tr)) * NV + v] = acc2[r];
        out[((size_t)b * NT + (48 + tr)) * NV + v] = acc3[r];
    }
}

extern "C" void kernel_launch(void* const* d_in, const int* in_sizes, int n_in,
                              void* d_out, int out_size, void* d_ws, size_t ws_size,
                              hipStream_t stream) {
    const int*   ids    = (const int*)  d_in[0];   // (B,T) int32
    const float* fslow  = (const float*)d_in[1];   // (V,)
    const float* ffast  = (const float*)d_in[2];   // (V,)
    const float* amp    = (const float*)d_in[3];   // (V,)
    const float* ph     = (const float*)d_in[4];   // (V,)
    const float* smix   = (const float*)d_in[5];   // (V,)
    const float* dslow  = (const float*)d_in[6];   // scalar
    const float* dfast  = (const float*)d_in[7];   // scalar
    const float* lsraw  = (const float*)d_in[8];   // scalar
    const float* lfraw  = (const float*)d_in[9];   // scalar
    // d_in[10] gate_filter, d_in[11] gate_bias: gate is mathematically identity
    const float* temp   = (const float*)d_in[12];  // scalar

    float* Cbuf = (float*)d_ws;                    // (B,T,V) f32 = 4 MB scratch
    float* out  = (float*)d_out;                   // (B,T,V) f32

    dim3 g1(NV / 256, NT, NB);
    WaveLM_cross_kernel<<<g1, 256, 0, stream>>>(ids, fslow, ffast, amp, ph, smix,
                                                dslow, dfast, Cbuf);

    // One wave per (batch, 16-wide v stripe): NB * (NV/16) = 1024 waves,
    // 8 waves per 256-thread block -> 128 blocks, exact.
    const int nWaves = NB * (NV / 16);
    const int blocks = (nWaves * 32) / 256;        // = 128
    WaveLM_scan_wmma_kernel<<<blocks, 256, 0, stream>>>(Cbuf, lsraw, lfraw, temp, out);
}